// GNN_74938589380918
// MI455X (gfx1250) — compile-verified
//
#include <hip/hip_runtime.h>
#include <math.h>

#define N_NODES 50000   // multiple of 16 -> no row guards in node GEMMs
#define N_EDGES 150000  // multiple of 16 -> only last 32-edge tile is partial

typedef __attribute__((ext_vector_type(2))) float    v2f;
typedef __attribute__((ext_vector_type(8))) float    v8f;
typedef __attribute__((ext_vector_type(4))) unsigned v4u;
typedef __attribute__((ext_vector_type(8))) int      v8i;
typedef __attribute__((ext_vector_type(4))) int      v4i;

// CDNA5 fp32 WMMA: D(16x16,f32) = A(16x4,f32) x B(4x16,f32) + C
__device__ __forceinline__ v8f wmma_f32x4(v2f a, v2f b, v8f c) {
  return __builtin_amdgcn_wmma_f32_16x16x4_f32(false, a, false, b, (short)0, c,
                                               false, false);
}

// ---------------------------------------------------------------------------
// Tensor Data Mover: DMA a 2D f32 tile [rows x cols] (row stride = `stride`
// elements) from global memory into LDS at byte offset lds_addr.
// D# layout per CDNA5 ISA ch.8 (group0 128b, group1 256b, groups 2/3 zero).
// This toolchain exposes the 6-arg builtin:
//   (uint32x4 g0, int32x8 g1, int32x4 g2, int32x4 g3, int32x8 gx, i32 cpol)
// ---------------------------------------------------------------------------
__device__ __forceinline__ void tdm_load_2d_f32(unsigned lds_addr,
                                                const float* gptr,
                                                unsigned rows, unsigned cols,
                                                unsigned stride) {
  unsigned long long ga = (unsigned long long)(size_t)gptr;
  v4u g0;
  g0.x = 1u;                                             // count=1 (valid D#)
  g0.y = lds_addr;                                       // lds_addr [63:32]
  g0.z = (unsigned)(ga & 0xFFFFFFFFu);                   // global_addr [95:64]
  g0.w = (unsigned)((ga >> 32) & 0x01FFFFFFu)            // global_addr [120:96]
       | (2u << 30);                                     // type=2 ("image")
  v8i g1;
  g1[0] = (int)(2u << 16);                               // data_size=4B; mask=0
  g1[1] = (int)((stride & 0xFFFFu) << 16);               // tensor_dim0 lo16
  g1[2] = (int)(((stride >> 16) & 0xFFFFu)               // tensor_dim0 hi16
       | ((rows & 0xFFFFu) << 16));                      // tensor_dim1 lo16
  g1[3] = (int)(((rows >> 16) & 0xFFFFu)                 // tensor_dim1 hi16
       | ((cols & 0xFFFFu) << 16));                      // tile_dim0
  g1[4] = (int)(rows & 0xFFFFu);                         // tile_dim1, tile_dim2=0
  g1[5] = (int)stride;                                   // dim0_stride lo32
  g1[6] = 0;                                             // dim0_stride hi16
  g1[7] = 0;                                             // dim1_stride (unused)
  v4i gz = {0, 0, 0, 0};                                 // groups 2/3: 2D tensor
  v8i gz8 = {0, 0, 0, 0, 0, 0, 0, 0};
  __builtin_amdgcn_tensor_load_to_lds(g0, g1, gz, gz, gz8, 0);
}

// ---------------------------------------------------------------------------
// Degree / normalization precompute
// ---------------------------------------------------------------------------
__global__ void gnn_deg_init(float* __restrict__ deg) {
  int i = blockIdx.x * blockDim.x + threadIdx.x;
  if (i < N_NODES) deg[i] = 1.0f;  // self loop
}

__global__ void gnn_deg_acc(const int* __restrict__ dst, float* __restrict__ deg) {
  int i = blockIdx.x * blockDim.x + threadIdx.x;
  if (i < N_EDGES) atomicAdd(&deg[dst[i]], 1.0f);
}

__global__ void gnn_deg_fin(const float* __restrict__ deg, float* __restrict__ dinv,
                            float* __restrict__ rcnt) {
  int i = blockIdx.x * blockDim.x + threadIdx.x;
  if (i < N_NODES) {
    float d = deg[i];          // >= 1 always (self loop)
    dinv[i] = rsqrtf(d);
    rcnt[i] = 1.0f / d;
  }
}

// ---------------------------------------------------------------------------
// fp32 WMMA GEMM with TDM-staged weights:
//   C[M x Nc] = A[M x K] * W[K x Nc], row major, M%16==0, K<=256, Nc=64<<s.
// One block = 8 row-tiles (16 rows each) x one 64-col tile. The K x 64 weight
// panel is DMA'd into LDS once per block by the Tensor Data Mover and shared
// by all 8 waves (8x less L2 weight traffic than per-wave loads).
// Fragment layouts per ISA 7.12.2:
//   A 16x4 : lane = m + 16*(k>>1), vgpr = k&1     (one b64 load)
//   B 4x16 : lane = n + 16*(k>>1), vgpr = k&1     (ds_load from staged panel)
//   C 16x16: vgpr r -> M = r + 8*(lane>>4), N = lane&15
// ---------------------------------------------------------------------------
__global__ __launch_bounds__(256) void gnn_gemm_wmma(
    const float* __restrict__ A, const float* __restrict__ W,
    float* __restrict__ C, int M, int K, int Nc, int colShift) {
  __shared__ float Bs[256 * 64];          // K<=256 panel of 64 cols (64 KB max)
  const int tid  = threadIdx.x;
  const int lane = tid & 31;
  const int wv   = tid >> 5;
  const int colTile  = blockIdx.x & ((1 << colShift) - 1);
  const int rowBlock = blockIdx.x >> colShift;
  const int rowTiles = M >> 4;
  const int rowTile  = rowBlock * 8 + wv;
  const int col0 = colTile << 6;

  if (wv == 0) {                          // one TDM op per block
    unsigned ldsAddr = (unsigned)(size_t)(void*)Bs;   // flat[31:0] = LDS offset
    tdm_load_2d_f32(ldsAddr, W + col0, (unsigned)K, 64u, (unsigned)Nc);
    __builtin_amdgcn_s_wait_tensorcnt(0);
  }
  __syncthreads();

  if (rowTile < rowTiles) {
    const int row0 = rowTile << 4;
    const int m16  = lane & 15;
    const int half = lane >> 4;
    const float* Arow = A + (size_t)(row0 + m16) * K;

    v8f acc[4] = {};
    for (int k0 = 0; k0 < K; k0 += 4) {
      const int kk = k0 + half * 2;
      const v2f a = *(const v2f*)(Arow + kk);           // global_load_b64
      const float* b0 = Bs + kk * 64 + m16;
#pragma unroll
      for (int t = 0; t < 4; ++t) {
        v2f b;
        b.x = b0[t * 16];                               // ds_load
        b.y = b0[64 + t * 16];
        acc[t] = wmma_f32x4(a, b, acc[t]);
      }
    }
#pragma unroll
    for (int t = 0; t < 4; ++t) {
#pragma unroll
      for (int r = 0; r < 8; ++r) {
        int row = row0 + r + half * 8;
        C[(size_t)row * Nc + col0 + t * 16 + m16] = acc[t][r];
      }
    }
  }
}

// ---------------------------------------------------------------------------
// Aggregation: agg = selfloop + scatter(edges), then mean + bias + BN + LReLU
// ---------------------------------------------------------------------------
__global__ void gnn_agg_self(const float* __restrict__ lin,
                             const float* __restrict__ dinv,
                             float* __restrict__ agg, int fshift, int total) {
  int idx = blockIdx.x * blockDim.x + threadIdx.x;
  if (idx >= total) return;
  int i = idx >> fshift;
  float di = dinv[i];
  agg[idx] = lin[idx] * di * di;   // also zero-initializes agg
}

__global__ void gnn_agg_edge(const float* __restrict__ lin,
                             const int* __restrict__ src,
                             const int* __restrict__ dst,
                             const float* __restrict__ dinv,
                             float* __restrict__ agg, int F) {
  int e = blockIdx.x;
  int s = src[e], d = dst[e];
  float nrm = dinv[s] * dinv[d];
  const float* ls = lin + (size_t)s * F;
  float* ad = agg + (size_t)d * F;
  for (int f = threadIdx.x; f < F; f += blockDim.x)
    atomicAdd(&ad[f], ls[f] * nrm);
}

__global__ void gnn_finish(const float* __restrict__ agg,
                           const float* __restrict__ rcnt,
                           const float* __restrict__ b, const float* __restrict__ g,
                           const float* __restrict__ be, float* __restrict__ out,
                           int fshift, int total) {
  int idx = blockIdx.x * blockDim.x + threadIdx.x;
  if (idx >= total) return;
  int i = idx >> fshift;
  int f = idx & ((1 << fshift) - 1);
  float v = agg[idx] * rcnt[i] + b[f];
  v = v * (g[f] * rsqrtf(1.0f + 1e-5f)) + be[f];
  out[idx] = v >= 0.0f ? v : 0.01f * v;
}

// ---------------------------------------------------------------------------
// Fused edge predictor: 32 edges / block, full MLP through LDS.
// Tail-tile edge indices are clamped; garbage rows masked at the pred store.
// ---------------------------------------------------------------------------
__global__ __launch_bounds__(256) void gnn_edge_mlp(
    const float* __restrict__ h, const int* __restrict__ src,
    const int* __restrict__ dst,
    const float* __restrict__ Wp1, const float* __restrict__ bp1,
    const float* __restrict__ Wp2, const float* __restrict__ bp2,
    const float* __restrict__ Wp3, const float* __restrict__ bp3,
    float* __restrict__ pred, int E) {
  __shared__ float e1s[32 * 256];
  __shared__ float e2s[32 * 128];
  const int tid  = threadIdx.x;
  const int lane = tid & 31;
  const int wv   = tid >> 5;
  const int rt   = wv >> 2;          // row tile (0..1): edges [rt*16, rt*16+16)
  const int m16  = lane & 15;
  const int half = lane >> 4;
  const long e0  = (long)blockIdx.x * 32;
  long er = e0 + rt * 16 + m16;
  if (er >= E) er = E - 1;           // clamp: keep loads in-bounds, no masking
  const int si = src[er];
  const int di = dst[er];
  const float* rA = h + (size_t)si * 512;
  const float* rB = h + (size_t)di * 512;

  // ---- stage 1: K = 1024 (512 from h[src], 512 from h[dst]), N = 256 ----
  {
    const int cb = (wv & 3) * 64;
    v8f acc[4] = {};
    for (int k0 = 0; k0 < 512; k0 += 4) {
      const int kk = k0 + half * 2;
      const v2f a = *(const v2f*)(rA + kk);
      const float* w0 = Wp1 + (size_t)kk * 256 + cb + m16;
#pragma unroll
      for (int t = 0; t < 4; ++t) {
        v2f b; b.x = w0[t * 16]; b.y = w0[256 + t * 16];
        acc[t] = wmma_f32x4(a, b, acc[t]);
      }
    }
    for (int k0 = 512; k0 < 1024; k0 += 4) {
      const int kk = k0 + half * 2;
      const v2f a = *(const v2f*)(rB + (kk - 512));
      const float* w0 = Wp1 + (size_t)kk * 256 + cb + m16;
#pragma unroll
      for (int t = 0; t < 4; ++t) {
        v2f b; b.x = w0[t * 16]; b.y = w0[256 + t * 16];
        acc[t] = wmma_f32x4(a, b, acc[t]);
      }
    }
#pragma unroll
    for (int t = 0; t < 4; ++t) {
#pragma unroll
      for (int r = 0; r < 8; ++r) {
        int row = rt * 16 + r + half * 8;
        int col = cb + t * 16 + m16;
        float v = acc[t][r] + bp1[col];
        e1s[row * 256 + col] = v > 0.0f ? v : 0.0f;
      }
    }
  }
  __syncthreads();

  // ---- stage 2: K = 256, N = 128 ----
  {
    const int cb = (wv & 3) * 32;
    v8f acc[2] = {};
    const float* Ar = e1s + (rt * 16 + m16) * 256;
    for (int k0 = 0; k0 < 256; k0 += 4) {
      const int kk = k0 + half * 2;
      const v2f a = *(const v2f*)(Ar + kk);               // ds_load_b64
      const float* w0 = Wp2 + (size_t)kk * 128 + cb + m16;
#pragma unroll
      for (int t = 0; t < 2; ++t) {
        v2f b; b.x = w0[t * 16]; b.y = w0[128 + t * 16];
        acc[t] = wmma_f32x4(a, b, acc[t]);
      }
    }
#pragma unroll
    for (int t = 0; t < 2; ++t) {
#pragma unroll
      for (int r = 0; r < 8; ++r) {
        int row = rt * 16 + r + half * 8;
        int col = cb + t * 16 + m16;
        float v = acc[t][r] + bp2[col];
        e2s[row * 128 + col] = v > 0.0f ? v : 0.0f;
      }
    }
  }
  __syncthreads();

  // ---- stage 3: K = 128, N = 1 + sigmoid ----
  if (tid < 32) {
    float s = bp3[0];
    const float* r2 = e2s + tid * 128;
#pragma unroll 8
    for (int k = 0; k < 128; ++k) s = fmaf(r2[k], Wp3[k], s);
    float p = 1.0f / (1.0f + __expf(-s));
    long ee = e0 + tid;
    if (ee < E) pred[ee] = p;
  }
}

// ---------------------------------------------------------------------------
extern "C" void kernel_launch(void* const* d_in, const int* in_sizes, int n_in,
                              void* d_out, int out_size, void* d_ws, size_t ws_size,
                              hipStream_t stream) {
  (void)in_sizes; (void)n_in; (void)out_size; (void)ws_size;
  const float* x   = (const float*)d_in[0];
  const int*   ei  = (const int*)d_in[1];
  const int*   src = ei;
  const int*   dst = ei + N_EDGES;
  const float* W1 = (const float*)d_in[2],  *b1 = (const float*)d_in[3];
  const float* g1 = (const float*)d_in[4],  *be1 = (const float*)d_in[5];
  const float* W2 = (const float*)d_in[6],  *b2 = (const float*)d_in[7];
  const float* g2 = (const float*)d_in[8],  *be2 = (const float*)d_in[9];
  const float* W3 = (const float*)d_in[10], *b3 = (const float*)d_in[11];
  const float* g3 = (const float*)d_in[12], *be3 = (const float*)d_in[13];
  const float* Wp1 = (const float*)d_in[14], *bp1 = (const float*)d_in[15];
  const float* Wp2 = (const float*)d_in[16], *bp2 = (const float*)d_in[17];
  const float* Wp3 = (const float*)d_in[18], *bp3 = (const float*)d_in[19];

  float* out  = (float*)d_out;
  float* h3   = out;                               // [50000 x 512]
  float* pred = out + (size_t)N_NODES * 512;       // [150000]
  float* agg  = out;  // h-region of d_out doubles as aggregation scratch

  float* ws   = (float*)d_ws;
  float* deg  = ws;
  float* dinv = ws + N_NODES;
  float* rcnt = ws + 2 * (size_t)N_NODES;
  float* lin  = ws + 3 * (size_t)N_NODES;          // [50000 x 512] max
  float* hbuf = lin + (size_t)N_NODES * 512;       // [50000 x 256] max

  // degree / normalization
  gnn_deg_init<<<(N_NODES + 255) / 256, 256, 0, stream>>>(deg);
  gnn_deg_acc<<<(N_EDGES + 255) / 256, 256, 0, stream>>>(dst, deg);
  gnn_deg_fin<<<(N_NODES + 255) / 256, 256, 0, stream>>>(deg, dinv, rcnt);

  auto layer = [&](const float* Ain, int K, int F, int fshift, int colShift,
                   const float* W, const float* b, const float* g,
                   const float* be, float* outH) {
    int rowTiles = N_NODES / 16;                   // 3125
    int rowBlocks = (rowTiles + 7) / 8;            // 391
    gnn_gemm_wmma<<<rowBlocks << colShift, 256, 0, stream>>>(Ain, W, lin, N_NODES,
                                                             K, F, colShift);
    int total = N_NODES * F;
    gnn_agg_self<<<(total + 255) / 256, 256, 0, stream>>>(lin, dinv, agg, fshift,
                                                          total);
    gnn_agg_edge<<<N_EDGES, (F < 256 ? F : 256), 0, stream>>>(lin, src, dst, dinv,
                                                              agg, F);
    gnn_finish<<<(total + 255) / 256, 256, 0, stream>>>(agg, rcnt, b, g, be, outH,
                                                        fshift, total);
  };

  layer(x,    64,  128, 7, 1, W1, b1, g1, be1, hbuf);
  layer(hbuf, 128, 256, 8, 2, W2, b2, g2, be2, hbuf);
  layer(hbuf, 256, 512, 9, 3, W3, b3, g3, be3, h3);  // finish in-place into d_out

  gnn_edge_mlp<<<(N_EDGES + 31) / 32, 256, 0, stream>>>(
      h3, src, dst, Wp1, bp1, Wp2, bp2, Wp3, bp3, pred, N_EDGES);
}